// CombinedAdvancedLoss_62637803045460
// MI455X (gfx1250) — compile-verified
//
#include <hip/hip_runtime.h>
#include <hip/hip_bf16.h>
#include <math.h>

typedef __attribute__((ext_vector_type(2))) float v2f;
typedef __attribute__((ext_vector_type(8))) float v8f;

// ---- workspace layout (in floats) ----
#define ACC_FOCAL    0
#define ACC_CONTRAST 1
#define ACC_CIRC     2
#define ACC_CONS     3            // 6 slots: i01,u01,i02,u02,i12,u12
#define FN_OFF       16           // normalized features: 1024*512 floats
#define SIM_OFF      (16 + 1024*512)  // sim matrix: 1024*1024 floats

#define HW 65536                  // 256*256
#define NPIX (32*HW)

// -------------------------------------------------------------------------
__global__ void zero_acc_kernel(float* __restrict__ ws) {
    if (threadIdx.x < 16) ws[threadIdx.x] = 0.0f;
}

// -------------------------------------------------------------------------
// focal loss: per pixel 8-class log-softmax + focal weighting, global sum
__global__ void focal_kernel(const float* __restrict__ logits,
                             const int* __restrict__ target,
                             float* __restrict__ ws) {
    float acc = 0.0f;
    for (int p = blockIdx.x * blockDim.x + threadIdx.x; p < NPIX;
         p += gridDim.x * blockDim.x) {
        int b  = p >> 16;
        int hw = p & (HW - 1);
        const float* base = logits + b * 8 * HW + hw;
        float x[8];
        float m = -3.0e38f;
#pragma unroll
        for (int c = 0; c < 8; ++c) { x[c] = base[c * HW]; m = fmaxf(m, x[c]); }
        float s = 0.0f;
#pragma unroll
        for (int c = 0; c < 8; ++c) s += expf(x[c] - m);
        float lse = m + logf(s);
        int   t   = target[p];
        float ce  = lse - x[t];
        float pt  = expf(-ce);
        float w   = 0.25f * (1.0f - pt) * (1.0f - pt);
        acc += w * ce;
    }
    __shared__ float sd[256];
    sd[threadIdx.x] = acc;
    __syncthreads();
    for (int st = blockDim.x >> 1; st > 0; st >>= 1) {
        if (threadIdx.x < st) sd[threadIdx.x] += sd[threadIdx.x + st];
        __syncthreads();
    }
    if (threadIdx.x == 0) atomicAdd(&ws[ACC_FOCAL], sd[0]);
}

// -------------------------------------------------------------------------
// row-wise L2 normalization of features (1024 x 512) into ws[FN_OFF..]
__global__ void normalize_kernel(const float* __restrict__ feat,
                                 float* __restrict__ ws) {
    int row = blockIdx.x;
    const float* f  = feat + row * 512;
    float*       fn = ws + FN_OFF + row * 512;
    float ss = 0.0f;
    for (int i = threadIdx.x; i < 512; i += blockDim.x) {
        float v = f[i];
        ss += v * v;
    }
    __shared__ float sd[256];
    sd[threadIdx.x] = ss;
    __syncthreads();
    for (int st = blockDim.x >> 1; st > 0; st >>= 1) {
        if (threadIdx.x < st) sd[threadIdx.x] += sd[threadIdx.x + st];
        __syncthreads();
    }
    __shared__ float rn;
    if (threadIdx.x == 0) rn = rsqrtf(sd[0]);
    __syncthreads();
    for (int i = threadIdx.x; i < 512; i += blockDim.x) fn[i] = f[i] * rn;
}

// -------------------------------------------------------------------------
// sim = fn @ fn^T via V_WMMA_F32_16X16X4_F32. One wave per 16x16 C tile.
// A 16x4 layout (ISA 7.12.2): lanes 0-15 hold M=lane, VGPR0=K0/K2(half),
// VGPR1=K1/K3. B 4x16 symmetric with N over lanes. Since B = fn^T, the
// per-lane B fetch is fn[colBase+lane%16][k0 + 2*(lane/16) + {0,1}] — the
// same addressing pattern as A, so both operands are 8-byte float2 loads.
__global__ void __launch_bounds__(32)
sim_wmma_kernel(const float* __restrict__ fn, float* __restrict__ sim) {
    const int lane = threadIdx.x;
    const int half = lane >> 4;     // 0: K pair {0,1}, 1: K pair {2,3}
    const int lr   = lane & 15;
    const int rowBase = blockIdx.y * 16;
    const int colBase = blockIdx.x * 16;

    const float* aptr = fn + (rowBase + lr) * 512 + 2 * half;
    const float* bptr = fn + (colBase + lr) * 512 + 2 * half;

    v8f c = {};
#pragma unroll 4
    for (int k0 = 0; k0 < 512; k0 += 4) {
        v2f a = *reinterpret_cast<const v2f*>(aptr + k0);
        v2f b = *reinterpret_cast<const v2f*>(bptr + k0);
        // (neg_a, A, neg_b, B, c_mod, C, reuse_a, reuse_b)
        c = __builtin_amdgcn_wmma_f32_16x16x4_f32(false, a, false, b,
                                                  (short)0, c, false, false);
    }
    // C layout: VGPR r holds row r (lanes 0-15) and row r+8 (lanes 16-31)
#pragma unroll
    for (int r = 0; r < 8; ++r) {
        sim[(rowBase + r + 8 * half) * 1024 + colBase + lr] = c[r];
    }
}

// -------------------------------------------------------------------------
// per-row logsumexp of sim/TEMP with diagonal masked, minus positive pair
__global__ void lse_kernel(const float* __restrict__ sim,
                           float* __restrict__ ws) {
    const float INV_T = 1.0f / 0.07f;
    int row = blockIdx.x;
    const float* srow = sim + row * 1024;

    float m = -3.0e38f, s = 0.0f;
    for (int c = threadIdx.x; c < 1024; c += blockDim.x) {
        float v = (c == row) ? -1e30f : srow[c] * INV_T;
        if (v > m) { s = s * expf(m - v) + 1.0f; m = v; }
        else       { s += expf(v - m); }
    }
    __shared__ float sm[256], sv[256];
    sm[threadIdx.x] = m;
    sv[threadIdx.x] = s;
    __syncthreads();
    for (int st = blockDim.x >> 1; st > 0; st >>= 1) {
        if (threadIdx.x < st) {
            float m1 = sm[threadIdx.x], s1 = sv[threadIdx.x];
            float m2 = sm[threadIdx.x + st], s2 = sv[threadIdx.x + st];
            float mm = fmaxf(m1, m2);
            sm[threadIdx.x] = mm;
            sv[threadIdx.x] = s1 * expf(m1 - mm) + s2 * expf(m2 - mm);
        }
        __syncthreads();
    }
    if (threadIdx.x == 0) {
        float lse = sm[0] + logf(sv[0]);
        int   pidx = (row + 512) & 1023;
        float pos  = srow[pidx] * INV_T;
        atomicAdd(&ws[ACC_CONTRAST], lse - pos);
    }
}

// -------------------------------------------------------------------------
// circularity: one block per image, channel 0 of masks[B,3,H,W]
__global__ void circ_kernel(const float* __restrict__ masks,
                            float* __restrict__ ws) {
    int b = blockIdx.x;
    const float* m = masks + b * 3 * HW;
    float area = 0.0f, ex = 0.0f, ey = 0.0f;
    for (int p = threadIdx.x; p < HW; p += blockDim.x) {
        float v = m[p];
        area += v;
        int r = p >> 8, cc = p & 255;
        if (r  < 255) ex += fabsf(m[p + 256] - v);
        if (cc < 255) ey += fabsf(m[p + 1]   - v);
    }
    __shared__ float sd[256];
    __shared__ float res[3];
    float vals[3] = {area, ex, ey};
    for (int k = 0; k < 3; ++k) {
        __syncthreads();
        sd[threadIdx.x] = vals[k];
        __syncthreads();
        for (int st = blockDim.x >> 1; st > 0; st >>= 1) {
            if (threadIdx.x < st) sd[threadIdx.x] += sd[threadIdx.x + st];
            __syncthreads();
        }
        if (threadIdx.x == 0) res[k] = sd[0];
    }
    __syncthreads();
    if (threadIdx.x == 0) {
        float A = res[0], per = res[1] + res[2];
        float pm = fmaxf(per, 1e-12f);
        float circ = 4.0f * 3.14159265358979323846f * A / (pm * pm);
        float li = (A > 0.0f && per > 0.0f) ? (circ - 1.0f) * (circ - 1.0f)
                                            : 0.0f;
        atomicAdd(&ws[ACC_CIRC], li);
    }
}

// -------------------------------------------------------------------------
// consensus: pairwise IoU sums over 3 methods, single pass over data
__global__ void consensus_kernel(const float* __restrict__ mp,
                                 float* __restrict__ ws) {
    const float* p0 = mp;
    const float* p1 = mp + NPIX;
    const float* p2 = mp + 2 * NPIX;
    float i01 = 0, u01 = 0, i02 = 0, u02 = 0, i12 = 0, u12 = 0;
    for (int idx = blockIdx.x * blockDim.x + threadIdx.x; idx < NPIX;
         idx += gridDim.x * blockDim.x) {
        float a = p0[idx], b = p1[idx], c = p2[idx];
        float ab = a * b, ac = a * c, bc = b * c;
        i01 += ab; u01 += a + b - ab;
        i02 += ac; u02 += a + c - ac;
        i12 += bc; u12 += b + c - bc;
    }
    __shared__ float sd[256];
    float vals[6] = {i01, u01, i02, u02, i12, u12};
    for (int k = 0; k < 6; ++k) {
        __syncthreads();
        sd[threadIdx.x] = vals[k];
        __syncthreads();
        for (int st = blockDim.x >> 1; st > 0; st >>= 1) {
            if (threadIdx.x < st) sd[threadIdx.x] += sd[threadIdx.x + st];
            __syncthreads();
        }
        if (threadIdx.x == 0) atomicAdd(&ws[ACC_CONS + k], sd[0]);
    }
}

// -------------------------------------------------------------------------
__global__ void finalize_kernel(const float* __restrict__ ws,
                                float* __restrict__ out) {
    if (threadIdx.x == 0 && blockIdx.x == 0) {
        float focal    = ws[ACC_FOCAL] / (float)NPIX;
        float contrast = ws[ACC_CONTRAST] / 1024.0f;
        float circ     = 0.1f * ws[ACC_CIRC] / 32.0f;
        float cons = 0.0f;
        for (int k = 0; k < 3; ++k) {
            float inter = ws[ACC_CONS + 2 * k];
            float uni   = ws[ACC_CONS + 2 * k + 1];
            float iou   = inter / (uni + 1e-6f);
            float r     = 0.6f - iou;
            cons += (r > 0.0f) ? r : 0.0f;
        }
        cons /= 3.0f;
        out[0] = 1.0f * focal + 0.5f * contrast + circ + 0.3f * cons;
    }
}

// -------------------------------------------------------------------------
extern "C" void kernel_launch(void* const* d_in, const int* in_sizes, int n_in,
                              void* d_out, int out_size, void* d_ws,
                              size_t ws_size, hipStream_t stream) {
    const float* logits   = (const float*)d_in[0];
    const int*   target   = (const int*)d_in[1];
    const float* features = (const float*)d_in[2];
    const float* masks    = (const float*)d_in[3];
    const float* mpred    = (const float*)d_in[4];
    float* ws  = (float*)d_ws;
    float* out = (float*)d_out;

    zero_acc_kernel<<<1, 32, 0, stream>>>(ws);
    focal_kernel<<<2048, 256, 0, stream>>>(logits, target, ws);
    normalize_kernel<<<1024, 256, 0, stream>>>(features, ws);
    sim_wmma_kernel<<<dim3(64, 64), 32, 0, stream>>>(ws + FN_OFF, ws + SIM_OFF);
    lse_kernel<<<1024, 256, 0, stream>>>(ws + SIM_OFF, ws);
    circ_kernel<<<32, 256, 0, stream>>>(masks, ws);
    consensus_kernel<<<2048, 256, 0, stream>>>(mpred, ws);
    finalize_kernel<<<1, 32, 0, stream>>>(ws, out);
}